// SDGraphEncoder_7945689498141
// MI455X (gfx1250) — compile-verified
//
#include <hip/hip_runtime.h>
#include <hip/hip_bf16.h>

typedef float v2f __attribute__((ext_vector_type(2)));
typedef float v8f __attribute__((ext_vector_type(8)));

__device__ __forceinline__ float lrelu(float x) { return x > 0.f ? x : 0.2f * x; }

// ---------------------------------------------------------------------------
// Wave-level fp32 GEMM, B transposed: C[M,N] = A[M,K] * B^T, B is N x K
// row-major. One wave (32 lanes) per 16x16 C tile, V_WMMA_F32_16X16X4_F32.
// Layouts per CDNA5 ISA 7.12.2:
//   A 16x4 : lanes 0-15 row=lane, K=(0,1) in (v0,v1); lanes 16-31 K=(2,3)
//   B 4x16 : lanes 0-15 col=lane, K=(0,1) in (v0,v1); lanes 16-31 K=(2,3)
//   D 16x16: VGPR v -> row tm+v (lanes 0-15) / tm+v+8 (lanes 16-31), col=lane%16
// K % 8 == 0. Two accumulator chains + unroll so loads stay in flight and
// WMMAs from independent chains can overlap.
// ---------------------------------------------------------------------------
__global__ void k_gemm_f32_bt(const float* __restrict__ A, int lda,
                              const float* __restrict__ B, int ldb,
                              float* __restrict__ C, int ldc,
                              int N, int K) {
  const int ntn  = N >> 4;
  const int tm   = (blockIdx.x / ntn) << 4;
  const int tn   = (blockIdx.x % ntn) << 4;
  const int lane = threadIdx.x;
  const int half = lane >> 4;   // 0: K pair (0,1), 1: K pair (2,3)
  const int l16  = lane & 15;

  const float* ap = A + (long)(tm + l16) * lda + (half << 1);
  const float* bp = B + (long)(tn + l16) * ldb + (half << 1);

  v8f acc0 = {}, acc1 = {};
#pragma unroll 4
  for (int k = 0; k < K; k += 8) {
    const v2f a0 = *(const v2f*)(ap);
    const v2f b0 = *(const v2f*)(bp);
    const v2f a1 = *(const v2f*)(ap + 4);
    const v2f b1 = *(const v2f*)(bp + 4);
    ap += 8; bp += 8;
    acc0 = __builtin_amdgcn_wmma_f32_16x16x4_f32(false, a0, false, b0,
                                                 (short)0, acc0, false, false);
    acc1 = __builtin_amdgcn_wmma_f32_16x16x4_f32(false, a1, false, b1,
                                                 (short)0, acc1, false, false);
  }
#pragma unroll
  for (int v = 0; v < 8; ++v) {
    const int row = tm + v + (half ? 8 : 0);
    C[(long)row * ldc + (tn + l16)] = acc0[v] + acc1[v];
  }
}

// ---------------------------------------------------------------------------
// d2s conv (1x3, no pad): dense_fea (8,128,32,64) -> conv1 (8,128,32,62)
// plus per-channel sum / sumsq for BN.  One block per (b, ochan).
// ---------------------------------------------------------------------------
__global__ void k_d2s_conv(const float* __restrict__ df, const float* __restrict__ w,
                           const float* __restrict__ bias,
                           float* __restrict__ conv1, float* __restrict__ stat1) {
  const int b = blockIdx.x >> 7;
  const int o = blockIdx.x & 127;
  const int tid = threadIdx.x;  // 64 threads
  float s = 0.f, s2 = 0.f;
  for (int e = tid; e < 32 * 62; e += 64) {
    const int n = e / 62, wo = e % 62;
    float acc = bias[o];
    for (int i = 0; i < 128; ++i) {
      const float* xp = df + ((b * 128 + i) * 2048) + n * 64 + wo;
      const float* wp = w + (o * 128 + i) * 3;
      acc += xp[0] * wp[0] + xp[1] * wp[1] + xp[2] * wp[2];
    }
    conv1[((b * 128 + o) * 32 + n) * 62 + wo] = acc;
    s += acc; s2 += acc * acc;
  }
  __shared__ float sh[128];
  sh[tid] = s; sh[64 + tid] = s2;
  __syncthreads();
  for (int st = 32; st; st >>= 1) {
    if (tid < st) { sh[tid] += sh[tid + st]; sh[64 + tid] += sh[64 + tid + st]; }
    __syncthreads();
  }
  if (tid == 0) { atomicAdd(&stat1[o], sh[0]); atomicAdd(&stat1[128 + o], sh[64]); }
}

// union_sparse (8,384,32): c<256 copy sparse_fea; else BN+leaky+max over width
__global__ void k_union_sparse(const float* __restrict__ sf, const float* __restrict__ conv1,
                               const float* __restrict__ stat1, const float* __restrict__ g,
                               const float* __restrict__ be, float* __restrict__ us) {
  const int idx = blockIdx.x * blockDim.x + threadIdx.x;
  if (idx >= 8 * 384 * 32) return;
  const int n = idx & 31;
  const int c = (idx >> 5) % 384;
  const int b = idx / (384 * 32);
  float v;
  if (c < 256) {
    v = sf[(b * 256 + c) * 32 + n];
  } else {
    const int cc = c - 256;
    const float cnt = 8.f * 32.f * 62.f;
    const float m  = stat1[cc] / cnt;
    const float vr = stat1[128 + cc] / cnt - m * m;
    const float sc = g[cc] * rsqrtf(vr + 1e-5f);
    float mx = -3.4e38f;
    const float* p = conv1 + ((b * 128 + cc) * 32 + n) * 62;
    for (int wv = 0; wv < 62; ++wv) mx = fmaxf(mx, lrelu(sc * (p[wv] - m) + be[cc]));
    v = mx;
  }
  us[idx] = v;
}

// xts (8,32,384) = transpose of union_sparse
__global__ void k_xts(const float* __restrict__ us, float* __restrict__ xts) {
  const int t = blockIdx.x * blockDim.x + threadIdx.x;
  if (t >= 8 * 32 * 384) return;
  const int c = t % 384, bn = t / 384, b = bn / 32, n = bn % 32;
  xts[t] = us[(b * 384 + c) * 32 + n];
}

// xtd (8,2048,384): union_dense transposed to point-major
__global__ void k_xtd(const float* __restrict__ df, const float* __restrict__ sf,
                      float* __restrict__ xtd) {
  const int t = blockIdx.x * blockDim.x + threadIdx.x;
  if (t >= 8 * 2048 * 384) return;
  const int c = t % 384, bp = t / 384, p = bp % 2048, b = bp / 2048;
  xtd[t] = (c < 128) ? df[(b * 128 + c) * 2048 + p]
                     : sf[(b * 256 + (c - 128)) * 32 + (p >> 6)];
}

__global__ void k_normd(const float* __restrict__ xtd, float* __restrict__ normd) {
  const int t = blockIdx.x * blockDim.x + threadIdx.x;
  if (t >= 8 * 2048) return;
  const float* p = xtd + (long)t * 384;
  float s = 0.f;
  for (int c = 0; c < 384; ++c) s += p[c] * p[c];
  normd[t] = s;
}

// top-10 smallest distances per row, ties -> lower index (matches top_k)
__global__ void k_topk10(const float* __restrict__ gram, const float* __restrict__ normd,
                         int b, int* __restrict__ idxd) {
  const int n = blockIdx.x * blockDim.x + threadIdx.x;
  if (n >= 2048) return;
  const float nn = normd[b * 2048 + n];
  float vals[10]; int ids[10];
#pragma unroll
  for (int i = 0; i < 10; ++i) { vals[i] = 3.4e38f; ids[i] = 0; }
  const float* gr = gram + (long)n * 2048;
  for (int m = 0; m < 2048; ++m) {
    const float d = nn + normd[b * 2048 + m] - 2.f * gr[m];
    if (d < vals[9]) {
      int p = 9;
      while (p > 0 && d < vals[p - 1]) { vals[p] = vals[p - 1]; ids[p] = ids[p - 1]; --p; }
      vals[p] = d; ids[p] = m;
    }
  }
#pragma unroll
  for (int i = 0; i < 10; ++i) idxd[(b * 2048 + n) * 10 + i] = ids[i];
}

// dense GCN BN stats: h = G1[idx] - G1[n] + G2[n] + bias, per (b,n) block
__global__ void k_hd_stats(const float* __restrict__ G1, const float* __restrict__ G2,
                           const int* __restrict__ idxd, const float* __restrict__ bias,
                           float* __restrict__ statd) {
  const int bn = blockIdx.x;          // 0..16383
  const int c  = threadIdx.x;         // 256
  const int b  = bn >> 11;
  const float base = G2[(long)bn * 256 + c] - G1[(long)bn * 256 + c] + bias[c];
  float s = 0.f, s2 = 0.f;
  for (int k = 0; k < 10; ++k) {
    const int m = idxd[bn * 10 + k];
    const float h = G1[((long)(b * 2048 + m)) * 256 + c] + base;
    s += h; s2 += h * h;
  }
  atomicAdd(&statd[c], s); atomicAdd(&statd[256 + c], s2);
}

__global__ void k_hd_apply(const float* __restrict__ G1, const float* __restrict__ G2,
                           const int* __restrict__ idxd, const float* __restrict__ bias,
                           const float* __restrict__ statd, const float* __restrict__ g,
                           const float* __restrict__ be, float* __restrict__ udout) {
  const int bn = blockIdx.x;
  const int c  = threadIdx.x;
  const int b  = bn >> 11, n = bn & 2047;
  const float cnt = 8.f * 2048.f * 10.f;
  const float m  = statd[c] / cnt;
  const float vr = statd[256 + c] / cnt - m * m;
  const float sc = g[c] * rsqrtf(vr + 1e-5f);
  const float base = G2[(long)bn * 256 + c] - G1[(long)bn * 256 + c] + bias[c];
  float mx = -3.4e38f;
  for (int k = 0; k < 10; ++k) {
    const int mm = idxd[bn * 10 + k];
    const float h = G1[((long)(b * 2048 + mm)) * 256 + c] + base;
    mx = fmaxf(mx, lrelu(sc * (h - m) + be[c]));
  }
  udout[((long)(b * 256 + c)) * 2048 + n] = mx;
}

// sparse GCN: 2 nearest among 32 (direct distances)
__global__ void k_top2(const float* __restrict__ xts, int* __restrict__ idxs) {
  const int t = blockIdx.x * blockDim.x + threadIdx.x;
  if (t >= 256) return;
  const int b = t >> 5, n = t & 31;
  const float* xn = xts + (b * 32 + n) * 384;
  float v0 = 3.4e38f, v1 = 3.4e38f; int i0 = 0, i1 = 0;
  for (int m = 0; m < 32; ++m) {
    const float* xm = xts + (b * 32 + m) * 384;
    float d = 0.f;
    for (int c = 0; c < 384; ++c) { const float df = xn[c] - xm[c]; d += df * df; }
    if (d < v0)      { v1 = v0; i1 = i0; v0 = d; i0 = m; }
    else if (d < v1) { v1 = d; i1 = m; }
  }
  idxs[t * 2] = i0; idxs[t * 2 + 1] = i1;
}

__global__ void k_hs_stats(const float* __restrict__ G1, const float* __restrict__ G2,
                           const int* __restrict__ idxs, const float* __restrict__ bias,
                           float* __restrict__ stats) {
  const int bn = blockIdx.x;       // 256
  const int c  = threadIdx.x;      // 512
  const int b  = bn >> 5;
  const float base = G2[bn * 512 + c] - G1[bn * 512 + c] + bias[c];
  float s = 0.f, s2 = 0.f;
  for (int k = 0; k < 2; ++k) {
    const int m = idxs[bn * 2 + k];
    const float h = G1[(b * 32 + m) * 512 + c] + base;
    s += h; s2 += h * h;
  }
  atomicAdd(&stats[c], s); atomicAdd(&stats[512 + c], s2);
}

__global__ void k_hs_apply(const float* __restrict__ G1, const float* __restrict__ G2,
                           const int* __restrict__ idxs, const float* __restrict__ bias,
                           const float* __restrict__ stats, const float* __restrict__ g,
                           const float* __restrict__ be, float* __restrict__ usout) {
  const int t = blockIdx.x * blockDim.x + threadIdx.x;
  if (t >= 8 * 512 * 32) return;
  const int n = t & 31, c = (t >> 5) & 511, b = t / (512 * 32);
  const float cnt = 512.f;
  const float m  = stats[c] / cnt;
  const float vr = stats[512 + c] / cnt - m * m;
  const float sc = g[c] * rsqrtf(vr + 1e-5f);
  const int bn = b * 32 + n;
  const float base = G2[bn * 512 + c] - G1[bn * 512 + c] + bias[c];
  float mx = -3.4e38f;
  for (int k = 0; k < 2; ++k) {
    const int mm = idxs[bn * 2 + k];
    mx = fmaxf(mx, lrelu(sc * (G1[(b * 32 + mm) * 512 + c] + base - m) + be[c]));
  }
  usout[t] = mx;   // (b,512,32)
}

// Farthest point sampling: one wave per batch. Emits fps_idx and stk_coor_s.
__global__ void k_fps(const float* __restrict__ coor, int* __restrict__ fpsidx,
                      float* __restrict__ out3) {
  const int b = blockIdx.x, n = threadIdx.x;  // 32 lanes
  const float* xb = coor + b * 32 * 64;
  float dist = 1e10f;
  int far = 0;
  for (int s = 0; s < 16; ++s) {
    if (n == 0) fpsidx[b * 16 + s] = far;
    out3[(b * 16 + s) * 64 + n]      = xb[far * 64 + n];
    out3[(b * 16 + s) * 64 + n + 32] = xb[far * 64 + n + 32];
    float d = 0.f;
    for (int c = 0; c < 64; ++c) { const float df = xb[n * 64 + c] - xb[far * 64 + c]; d += df * df; }
    dist = fminf(dist, d);
    float best = dist; int bi = n;
    for (int off = 16; off; off >>= 1) {
      const float ob = __shfl_xor(best, off, 32);
      const int   oi = __shfl_xor(bi, off, 32);
      if (ob > best || (ob == best && oi < bi)) { best = ob; bi = oi; }
    }
    far = bi;
  }
}

__global__ void k_gather_sp(const float* __restrict__ usout, const int* __restrict__ fpsidx,
                            float* __restrict__ out1) {
  const int t = blockIdx.x * blockDim.x + threadIdx.x;
  if (t >= 8 * 512 * 16) return;
  const int j = t & 15, c = (t >> 4) & 511, b = t / (512 * 16);
  out1[t] = usout[(b * 512 + c) * 32 + fpsidx[b * 16 + j]];
}

__global__ void k_build_dnin(const float* __restrict__ udout, const int* __restrict__ fpsidx,
                             float* __restrict__ dnin) {
  const int t = blockIdx.x * blockDim.x + threadIdx.x;
  if (t >= 8 * 256 * 16 * 64) return;
  const int w = t & 63, j = (t >> 6) & 15, c = (t >> 10) & 255, b = t >> 18;
  dnin[t] = udout[((long)(b * 256 + c)) * 2048 + fpsidx[b * 16 + j] * 64 + w];
}

// dcv conv (1x3, pad 1, stride (1,2)) + BN stats; one wave per (b,o,j)
__global__ void k_dcv_conv(const float* __restrict__ dnin, const float* __restrict__ w,
                           const float* __restrict__ bias, float* __restrict__ convc,
                           float* __restrict__ statc) {
  const int blk = blockIdx.x;  // 8*256*16
  const int j = blk & 15, o = (blk >> 4) & 255, b = blk >> 12;
  const int wo = threadIdx.x;  // 32
  const int base = 2 * wo - 1;
  float acc = bias[o];
  for (int i = 0; i < 256; ++i) {
    const float* xp = dnin + ((b * 256 + i) * 16 + j) * 64;
    const float* wp = w + (o * 256 + i) * 3;
    const float x0 = (base >= 0)     ? xp[base]     : 0.f;
    const float x1 =                   xp[base + 1];
    const float x2 = (base + 2 < 64) ? xp[base + 2] : 0.f;
    acc += x0 * wp[0] + x1 * wp[1] + x2 * wp[2];
  }
  convc[((b * 256 + o) * 16 + j) * 32 + wo] = acc;
  float s = acc, s2 = acc * acc;
  for (int off = 16; off; off >>= 1) { s += __shfl_xor(s, off, 32); s2 += __shfl_xor(s2, off, 32); }
  if (wo == 0) { atomicAdd(&statc[o], s); atomicAdd(&statc[256 + o], s2); }
}

__global__ void k_dcv_apply(const float* __restrict__ convc, const float* __restrict__ statc,
                            const float* __restrict__ g, const float* __restrict__ be,
                            float* __restrict__ out2) {
  const int t = blockIdx.x * blockDim.x + threadIdx.x;
  if (t >= 8 * 256 * 512) return;
  const int o = (t >> 9) & 255;
  const float cnt = 8.f * 16.f * 32.f;
  const float m  = statc[o] / cnt;
  const float vr = statc[256 + o] / cnt - m * m;
  const float sc = g[o] * rsqrtf(vr + 1e-5f);
  out2[t] = lrelu(sc * (convc[t] - m) + be[o]);
}

// ---------------------------------------------------------------------------
extern "C" void kernel_launch(void* const* d_in, const int* in_sizes, int n_in,
                              void* d_out, int out_size, void* d_ws, size_t ws_size,
                              hipStream_t stream) {
  const float* sparse_fea = (const float*)d_in[0];   // (8,256,32)
  const float* dense_fea  = (const float*)d_in[1];   // (8,128,2048)
  const float* stk_coor   = (const float*)d_in[2];   // (8,32,64)
  const float* d2s_w  = (const float*)d_in[3];
  const float* d2s_b  = (const float*)d_in[4];
  const float* d2s_g  = (const float*)d_in[5];
  const float* d2s_be = (const float*)d_in[6];
  const float* gsp_w  = (const float*)d_in[7];       // (512,768)
  const float* gsp_b  = (const float*)d_in[8];
  const float* gsp_g  = (const float*)d_in[9];
  const float* gsp_be = (const float*)d_in[10];
  const float* gdn_w  = (const float*)d_in[11];      // (256,768)
  const float* gdn_b  = (const float*)d_in[12];
  const float* gdn_g  = (const float*)d_in[13];
  const float* gdn_be = (const float*)d_in[14];
  const float* dcv_w  = (const float*)d_in[15];
  const float* dcv_b  = (const float*)d_in[16];
  const float* dcv_g  = (const float*)d_in[17];
  const float* dcv_be = (const float*)d_in[18];

  float* out1 = (float*)d_out;                       // sp  (8,512,16)
  float* out2 = out1 + 65536;                        // dn  (8,256,512)
  float* out3 = out2 + 1048576;                      // stk_coor_s (8,16,64)

  // workspace carve-up
  float* f = (float*)d_ws;
  size_t o = 0;
  float* conv1  = f + o; o += 2031616;               // (8,128,32,62)
  float* stat1  = f + o; o += 256;
  float* us     = f + o; o += 98304;                 // (8,384,32)
  float* xts    = f + o; o += 98304;                 // (8,32,384)
  float* xtd    = f + o; o += 6291456;               // (8,2048,384)
  float* normd  = f + o; o += 16384;
  float* gram   = f + o; o += 4194304;               // (2048,2048) reused per batch
  float* G1d    = f + o; o += 4194304;               // (16384,256)
  float* G2d    = f + o; o += 4194304;
  float* statd  = f + o; o += 512;
  float* udout  = f + o; o += 4194304;               // (8,256,2048)
  float* G1s    = f + o; o += 131072;                // (256,512)
  float* G2s    = f + o; o += 131072;
  float* stats_ = f + o; o += 1024;
  float* usout  = f + o; o += 131072;                // (8,512,32)
  float* dnin   = f + o; o += 2097152;               // (8,256,16,64)
  float* convc  = f + o; o += 1048576;               // (8,256,16,32)
  float* statc  = f + o; o += 512;
  int* idxd   = (int*)(f + o); o += 163840;          // (8,2048,10)
  int* idxs   = (int*)(f + o); o += 512;             // (8,32,2)
  int* fpsidx = (int*)(f + o); o += 128;             // (8,16)

  hipMemsetAsync(stat1, 0, 256 * sizeof(float), stream);
  hipMemsetAsync(statd, 0, 512 * sizeof(float), stream);
  hipMemsetAsync(stats_, 0, 1024 * sizeof(float), stream);
  hipMemsetAsync(statc, 0, 512 * sizeof(float), stream);

  // --- d2s branch -> union_sparse ---
  k_d2s_conv<<<1024, 64, 0, stream>>>(dense_fea, d2s_w, d2s_b, conv1, stat1);
  k_union_sparse<<<(98304 + 255) / 256, 256, 0, stream>>>(sparse_fea, conv1, stat1,
                                                          d2s_g, d2s_be, us);
  k_xts<<<(98304 + 255) / 256, 256, 0, stream>>>(us, xts);
  k_xtd<<<(6291456 + 255) / 256, 256, 0, stream>>>(dense_fea, sparse_fea, xtd);
  k_normd<<<(16384 + 255) / 256, 256, 0, stream>>>(xtd, normd);

  // --- dense GCN: Gram (WMMA, C = X X^T) + top-10 per batch ---
  for (int b = 0; b < 8; ++b) {
    const float* xb = xtd + (long)b * 2048 * 384;
    k_gemm_f32_bt<<<128 * 128, 32, 0, stream>>>(xb, 384, xb, 384,
                                                gram, 2048, 2048, 384);
    k_topk10<<<8, 256, 0, stream>>>(gram, normd, b, idxd);
  }
  // G1d/G2d (WMMA): (16384,384) x (256,384)^T
  k_gemm_f32_bt<<<1024 * 16, 32, 0, stream>>>(xtd, 384, gdn_w,       768, G1d, 256,
                                              256, 384);
  k_gemm_f32_bt<<<1024 * 16, 32, 0, stream>>>(xtd, 384, gdn_w + 384, 768, G2d, 256,
                                              256, 384);
  k_hd_stats<<<16384, 256, 0, stream>>>(G1d, G2d, idxd, gdn_b, statd);
  k_hd_apply<<<16384, 256, 0, stream>>>(G1d, G2d, idxd, gdn_b, statd, gdn_g, gdn_be, udout);

  // --- sparse GCN: G1s/G2s (WMMA) + top-2 ---
  k_gemm_f32_bt<<<16 * 32, 32, 0, stream>>>(xts, 384, gsp_w,       768, G1s, 512,
                                            512, 384);
  k_gemm_f32_bt<<<16 * 32, 32, 0, stream>>>(xts, 384, gsp_w + 384, 768, G2s, 512,
                                            512, 384);
  k_top2<<<1, 256, 0, stream>>>(xts, idxs);
  k_hs_stats<<<256, 512, 0, stream>>>(G1s, G2s, idxs, gsp_b, stats_);
  k_hs_apply<<<(131072 + 255) / 256, 256, 0, stream>>>(G1s, G2s, idxs, gsp_b, stats_,
                                                       gsp_g, gsp_be, usout);

  // --- FPS + gathers + final conv ---
  k_fps<<<8, 32, 0, stream>>>(stk_coor, fpsidx, out3);
  k_gather_sp<<<(65536 + 255) / 256, 256, 0, stream>>>(usout, fpsidx, out1);
  k_build_dnin<<<(2097152 + 255) / 256, 256, 0, stream>>>(udout, fpsidx, dnin);
  k_dcv_conv<<<8 * 256 * 16, 32, 0, stream>>>(dnin, dcv_w, dcv_b, convc, statc);
  k_dcv_apply<<<(1048576 + 255) / 256, 256, 0, stream>>>(convc, statc, dcv_g, dcv_be, out2);
  (void)in_sizes; (void)n_in; (void)out_size; (void)ws_size;
}